// MessagePassing_45887430590541
// MI455X (gfx1250) — compile-verified
//
#include <hip/hip_runtime.h>

// MessagePassing scatter-add: out[dst[e], :] += x[src[e], :]
//   N = 100000 nodes, D = 128 f32 features, E = 1600000 edges.
//
// MI455X reasoning (compile-only; argued from HW specs):
//   * x (51.2 MB) + out (51.2 MB) fit together in the 192 MB L2, so all
//     gather/accumulate traffic is L2-resident; HBM moves only ~115 MB
//     compulsory (~5 us at 23.3 TB/s).
//   * A direct per-edge scatter needs E*128 = 204.8M f32 L2 RMW atomics.
//     Atomic units process ~4 B/op vs. 64B+ per load -- atomics would
//     dominate runtime by >10x over the 819 MB gather. So we invert the
//     scatter into a gather: build CSR (dst -> src list) with ~3.2M cheap
//     u32 atomics, then one wave32 per node accumulates its in-neighbors in
//     VGPRs and writes each out row once (plain global_store_b128).
//   * wave32 accumulate: lanes load 32 src ids coalesced, broadcast via
//     v_readlane (index becomes an SGPR -> saddr-form global_load_b128
//     gathers), lane l owns columns [4l,4l+4) as a float4 accumulator.
//   * No matmul structure -> WMMA/SWMMAC not applicable; the CDNA5-specific
//     paths used are wave32 readlane, saddr b128 gathers, NT temporal hints
//     on single-pass streams, and device-scope non-returning atomics in the
//     fallback.

#define N_NODES 100000
#define D_FEAT  128
#define N_EDGES 1600000

// ---------------------------------------------------------------------------
// CSR build pass 0: zero the per-node counters (d_ws is poisoned by harness).
// ---------------------------------------------------------------------------
__global__ __launch_bounds__(256) void zero_counts_kernel(int* __restrict__ counts)
{
    int i = blockIdx.x * blockDim.x + threadIdx.x;
    if (i < N_NODES) counts[i] = 0;
}

// ---------------------------------------------------------------------------
// CSR build pass 1: histogram of destination nodes. 1.6M u32 atomics (cheap).
// ---------------------------------------------------------------------------
__global__ __launch_bounds__(256) void hist_kernel(const int* __restrict__ ei,
                                                   int* __restrict__ counts)
{
    int e = blockIdx.x * blockDim.x + threadIdx.x;
    if (e < N_EDGES) {
        int d = __builtin_nontemporal_load(ei + N_EDGES + e); // dst row
        atomicAdd(&counts[d], 1);  // global_atomic_add_u32, no return
    }
}

// ---------------------------------------------------------------------------
// CSR build pass 2: exclusive prefix sum over 100K counts. Single 1024-thread
// block, LDS Hillis-Steele per 1024-chunk with a running carry (~98 chunks;
// negligible vs. main pass). Writes offsets[N+1] and a working copy cursors[N].
// ---------------------------------------------------------------------------
__global__ __launch_bounds__(1024) void scan_kernel(const int* __restrict__ counts,
                                                    int* __restrict__ offsets,
                                                    int* __restrict__ cursors)
{
    __shared__ int smem[1024];
    __shared__ int s_carry;
    const int tid = threadIdx.x;
    if (tid == 0) s_carry = 0;
    __syncthreads();

    for (int base = 0; base < N_NODES; base += 1024) {
        int i = base + tid;
        int v = (i < N_NODES) ? counts[i] : 0;
        smem[tid] = v;
        __syncthreads();
        #pragma unroll
        for (int off = 1; off < 1024; off <<= 1) {
            int t = (tid >= off) ? smem[tid - off] : 0;
            __syncthreads();
            smem[tid] += t;
            __syncthreads();
        }
        int incl  = smem[tid];      // inclusive scan within chunk
        int carry = s_carry;        // everyone reads carry before it changes
        int total = smem[1023];     // chunk total
        __syncthreads();
        if (i < N_NODES) {
            int excl = carry + incl - v;
            offsets[i] = excl;
            cursors[i] = excl;
        }
        if (tid == 1023) s_carry = carry + total;
        __syncthreads();
    }
    if (tid == 0) offsets[N_NODES] = s_carry;   // == N_EDGES
}

// ---------------------------------------------------------------------------
// CSR build pass 3: bucket fill. pos = cursor[dst]++ ; srcs[pos] = src.
// 1.6M returning u32 atomics + 1.6M scattered 4B stores.
// ---------------------------------------------------------------------------
__global__ __launch_bounds__(256) void fill_kernel(const int* __restrict__ ei,
                                                   int* __restrict__ cursors,
                                                   int* __restrict__ srcs)
{
    int e = blockIdx.x * blockDim.x + threadIdx.x;
    if (e < N_EDGES) {
        int s = __builtin_nontemporal_load(ei + e);
        int d = __builtin_nontemporal_load(ei + N_EDGES + e);
        int pos = atomicAdd(&cursors[d], 1);   // global_atomic_add_u32 w/ return
        srcs[pos] = s;
    }
}

// ---------------------------------------------------------------------------
// Main pass: one wave32 per node. Lane l owns columns [4l, 4l+4) (float4 acc).
// src ids are fetched 32-at-a-time coalesced, then broadcast with v_readlane
// so the row gather uses a scalar base (saddr global_load_b128). Each output
// row is written exactly once -> zero f32 atomics, no output pre-zeroing.
// ---------------------------------------------------------------------------
__global__ __launch_bounds__(256) void MessagePassing_45887430590541_kernel(
    const float* __restrict__ x,
    const int*   __restrict__ offsets,
    const int*   __restrict__ srcs,
    float*       __restrict__ out)
{
    long long idx = (long long)blockIdx.x * blockDim.x + threadIdx.x;
    int node = (int)(idx >> 5);
    int lane = (int)(idx & 31);
    if (node >= N_NODES) return;     // wave-uniform: whole wave stays active

    int beg = offsets[node];
    int end = offsets[node + 1];

    float4 acc = make_float4(0.0f, 0.0f, 0.0f, 0.0f);

    for (int j0 = beg; j0 < end; j0 += 32) {
        int cnt = end - j0; if (cnt > 32) cnt = 32;
        // coalesced load of up to 32 src ids (one per lane)
        int sv = (j0 + lane < end) ? __builtin_nontemporal_load(srcs + j0 + lane) : 0;
        for (int k = 0; k < cnt; ++k) {
            int s = __builtin_amdgcn_readlane(sv, k);     // SGPR src id
            const float4 v =
                *(const float4*)(x + (size_t)s * D_FEAT + (size_t)lane * 4);
            acc.x += v.x; acc.y += v.y; acc.z += v.z; acc.w += v.w;
        }
    }

    *(float4*)(out + (size_t)node * D_FEAT + (size_t)lane * 4) = acc;
}

// ---------------------------------------------------------------------------
// Fallback (only if d_ws is too small for CSR): direct per-edge scatter with
// HW device-scope non-returning f32 atomics. One VGPR address pair serves all
// four adds via offset: immediates.
// ---------------------------------------------------------------------------
__device__ __forceinline__ void gatomic_add_f32(float* p, float v, int off) {
    asm volatile("global_atomic_add_f32 %0, %1, off offset:%c2 scope:SCOPE_DEV"
                 :
                 : "v"(p), "v"(v), "i"(off)
                 : "memory");
}

__global__ __launch_bounds__(256) void zero_out_kernel(float4* __restrict__ out4)
{
    int i = blockIdx.x * blockDim.x + threadIdx.x;
    if (i < (N_NODES * D_FEAT) / 4)
        out4[i] = make_float4(0.0f, 0.0f, 0.0f, 0.0f);
}

__global__ __launch_bounds__(256) void scatter_atomic_kernel(
    const float* __restrict__ x,
    const int*   __restrict__ ei,
    float*       __restrict__ out)
{
    long long idx = (long long)blockIdx.x * blockDim.x + threadIdx.x;
    int e    = (int)(idx >> 5);
    int lane = (int)(idx & 31);
    if (e >= N_EDGES) return;
    int src = __builtin_nontemporal_load(ei + e);
    int dst = __builtin_nontemporal_load(ei + N_EDGES + e);
    const float4 v = *(const float4*)(x + (size_t)src * D_FEAT + (size_t)lane * 4);
    float* p = out + (size_t)dst * D_FEAT + (size_t)lane * 4;
    gatomic_add_f32(p, v.x, 0);
    gatomic_add_f32(p, v.y, 4);
    gatomic_add_f32(p, v.z, 8);
    gatomic_add_f32(p, v.w, 12);
}

// ---------------------------------------------------------------------------
extern "C" void kernel_launch(void* const* d_in, const int* in_sizes, int n_in,
                              void* d_out, int out_size, void* d_ws, size_t ws_size,
                              hipStream_t stream)
{
    const float* x  = (const float*)d_in[0];   // [N, D] f32
    const int*   ei = (const int*)d_in[1];     // [2, E] i32
    float*       out = (float*)d_out;          // [N, D] f32

    // Workspace layout (ints): counts[N] | offsets[N+1] | cursors[N] | srcs[E]
    const size_t needed = (size_t)(3 * N_NODES + 1 + N_EDGES) * sizeof(int);

    if (ws_size >= needed) {
        int* counts  = (int*)d_ws;
        int* offsets = counts + N_NODES;
        int* cursors = offsets + (N_NODES + 1);
        int* srcs    = cursors + N_NODES;

        zero_counts_kernel<<<(N_NODES + 255) / 256, 256, 0, stream>>>(counts);
        hist_kernel<<<N_EDGES / 256, 256, 0, stream>>>(ei, counts);
        scan_kernel<<<1, 1024, 0, stream>>>(counts, offsets, cursors);
        fill_kernel<<<N_EDGES / 256, 256, 0, stream>>>(ei, cursors, srcs);
        // N * 32 lanes = 3,200,000 threads = exactly 12,500 blocks of 256.
        MessagePassing_45887430590541_kernel<<<(N_NODES * 32) / 256, 256, 0, stream>>>(
            x, offsets, srcs, out);
    } else {
        // Fallback: direct atomic scatter (d_ws too small for CSR build).
        zero_out_kernel<<<((N_NODES * D_FEAT) / 4 + 255) / 256, 256, 0, stream>>>(
            (float4*)out);
        scatter_atomic_kernel<<<(N_EDGES * 32) / 256, 256, 0, stream>>>(x, ei, out);
    }
}